// Create_11364483465561
// MI455X (gfx1250) — compile-verified
//
#include <hip/hip_runtime.h>

// Problem constants (fixed by the reference).
#define FOCK_D   4096
#define ROWS     (2 * FOCK_D)   // 8192 output rows
#define BATCH    2048           // columns (batch)
#define CVEC     8              // complex elements per thread (256 thr * 8 = 2048)

typedef float f32x4 __attribute__((ext_vector_type(4)));

// out[row, col] (complex64, interleaved re/im):
//   row 0,1      -> 0
//   row i (i>=2) -> x[i-2, col] * sqrt((i-2)/2 + 1)
//
// One 256-thread block (8 wave32 waves) per output row:
//  - row index is scalar -> uniform branch, saddr-based addressing
//  - 4x global_load_b128 (RT: 128 MiB of inputs stay resident in 192 MB L2)
//  - 4x global_store_b128 th:NT (write-once output must not evict inputs)
//  - v_sqrt_f32 (single TRANS op) for the row scale
__global__ __launch_bounds__(256)
void fock_create_shift_scale(const float* __restrict__ x_re,
                             const float* __restrict__ x_im,
                             float* __restrict__ out)
{
    const unsigned row = blockIdx.x;                 // scalar per block
    const unsigned col = threadIdx.x * CVEC;         // 0..2040, step 8

    float* o = out + ((size_t)row * BATCH + col) * 2;

    if (row < 2) {                                   // uniform branch (blocks 0,1)
        f32x4 z = (f32x4)0.0f;
        __builtin_nontemporal_store(z, (f32x4*)(o + 0));
        __builtin_nontemporal_store(z, (f32x4*)(o + 4));
        __builtin_nontemporal_store(z, (f32x4*)(o + 8));
        __builtin_nontemporal_store(z, (f32x4*)(o + 12));
        return;
    }

    const unsigned k = row - 2;                      // source row (scalar)
    const float    s = __builtin_amdgcn_sqrtf((float)((k >> 1) + 1));

    const float* rp = x_re + (size_t)k * BATCH + col;
    const float* ip = x_im + (size_t)k * BATCH + col;

    f32x4 r0 = *(const f32x4*)(rp + 0);              // global_load_b128
    f32x4 r1 = *(const f32x4*)(rp + 4);
    f32x4 i0 = *(const f32x4*)(ip + 0);
    f32x4 i1 = *(const f32x4*)(ip + 4);

    // Interleave re/im into four contiguous 16B packets, scaled.
    f32x4 o0 = { r0.x * s, i0.x * s, r0.y * s, i0.y * s };
    f32x4 o1 = { r0.z * s, i0.z * s, r0.w * s, i0.w * s };
    f32x4 o2 = { r1.x * s, i1.x * s, r1.y * s, i1.y * s };
    f32x4 o3 = { r1.z * s, i1.z * s, r1.w * s, i1.w * s };

    __builtin_nontemporal_store(o0, (f32x4*)(o + 0));   // global_store_b128 th:NT
    __builtin_nontemporal_store(o1, (f32x4*)(o + 4));
    __builtin_nontemporal_store(o2, (f32x4*)(o + 8));
    __builtin_nontemporal_store(o3, (f32x4*)(o + 12));
}

extern "C" void kernel_launch(void* const* d_in, const int* in_sizes, int n_in,
                              void* d_out, int out_size, void* d_ws, size_t ws_size,
                              hipStream_t stream)
{
    (void)in_sizes; (void)n_in; (void)out_size; (void)d_ws; (void)ws_size;

    const float* x_re = (const float*)d_in[0];
    const float* x_im = (const float*)d_in[1];
    float*       out  = (float*)d_out;

    // One block per output row; each block writes 2048 complex = 16 KiB.
    fock_create_shift_scale<<<ROWS, 256, 0, stream>>>(x_re, x_im, out);
}